// SimpleRNN_27479200760163
// MI455X (gfx1250) — compile-verified
//
#include <hip/hip_runtime.h>
#include <math.h>

// ---- problem constants (from reference) ----
#define T_STEPS 512
#define BATCH   64
#define DIN     256
#define HID     1024
#define ODIM    256

typedef float v2f __attribute__((ext_vector_type(2)));
typedef float v8f __attribute__((ext_vector_type(8)));

// D = A(16x4 f32) * B(4x16 f32) + C(16x16 f32), full-wave op.
__device__ __forceinline__ v8f wmma4_f32(v2f a, v2f b, v8f c) {
    // 8 args: (neg_a, A, neg_b, B, c_mod, C, reuse_a, reuse_b)
    return __builtin_amdgcn_wmma_f32_16x16x4_f32(false, a, false, b, (short)0, c,
                                                 false, false);
}

__global__ void rnn_zero_kernel(float* __restrict__ p, int n) {
    int i = blockIdx.x * blockDim.x + threadIdx.x;
    if (i < n) p[i] = 0.0f;
}

// One recurrence step: hout = tanh(xt @ Wx + hin @ Wh + b)
// Grid: 8 blocks x 256 threads (8 waves). Wave w of block bk owns column tile
// n0 = (bk*8+w)*16 and all 4 row tiles (B=64 rows). B-fragment (weights) is
// loaded once per k-step and shared across the 4 row-tile WMMAs.
__global__ void __launch_bounds__(256)
rnn_step_kernel(const float* __restrict__ xt,    // (64, 256)   row-major
                const float* __restrict__ Wx,    // (256, 1024) row-major
                const float* __restrict__ Wh,    // (1024,1024) row-major
                const float* __restrict__ bias,  // (1024)
                const float* __restrict__ hin,   // (64, 1024)
                float* __restrict__ hout)        // (64, 1024)
{
    const int wave = threadIdx.x >> 5;
    const int lane = threadIdx.x & 31;
    const int m16  = lane & 15;            // row (A) / col (B,C/D) within tile
    const int kk   = (lane >> 4) << 1;     // K sub-offset: 0 or 2
    const int n0   = (blockIdx.x * 8 + wave) * 16;   // output column base

    v8f zero = {0.f, 0.f, 0.f, 0.f, 0.f, 0.f, 0.f, 0.f};
    v8f acc[4];
#pragma unroll
    for (int r = 0; r < 4; ++r) acc[r] = zero;

    // ---- K segment 1: xt @ Wx (K = 256) ----
#pragma unroll 4
    for (int k0 = 0; k0 < DIN; k0 += 4) {
        v2f bf;
        bf.x = Wx[(k0 + kk) * HID + n0 + m16];
        bf.y = Wx[(k0 + kk + 1) * HID + n0 + m16];
#pragma unroll
        for (int r = 0; r < 4; ++r) {
            v2f a = *(const v2f*)(xt + (r * 16 + m16) * DIN + k0 + kk);
            acc[r] = wmma4_f32(a, bf, acc[r]);
        }
    }

    // ---- K segment 2: hin @ Wh (K = 1024) ----
#pragma unroll 4
    for (int k0 = 0; k0 < HID; k0 += 4) {
        v2f bf;
        bf.x = Wh[(k0 + kk) * HID + n0 + m16];
        bf.y = Wh[(k0 + kk + 1) * HID + n0 + m16];
#pragma unroll
        for (int r = 0; r < 4; ++r) {
            v2f a = *(const v2f*)(hin + (r * 16 + m16) * HID + k0 + kk);
            acc[r] = wmma4_f32(a, bf, acc[r]);
        }
    }

    // ---- epilogue: bias + tanh + store ----
    // C/D layout: vgpr v, lane<16 -> row v, col lane ; lane>=16 -> row 8+v, col lane-16
    const float bv   = bias[n0 + m16];
    const int   row0 = (lane >> 4) * 8;
#pragma unroll
    for (int r = 0; r < 4; ++r) {
#pragma unroll
        for (int v = 0; v < 8; ++v) {
            float val = tanhf(acc[r][v] + bv);
            hout[(r * 16 + row0 + v) * HID + n0 + m16] = val;
        }
    }
}

// y = h @ W2 + b2   (64x1024)@(1024x256) -> (64,256)
// Grid: 2 blocks x 256 threads (16 waves total; 16 column tiles of 16).
__global__ void __launch_bounds__(256)
rnn_out_kernel(const float* __restrict__ h,    // (64, 1024)
               const float* __restrict__ W2,   // (1024, 256)
               const float* __restrict__ b2,   // (256)
               float* __restrict__ y)          // (64, 256)
{
    const int wave = threadIdx.x >> 5;
    const int lane = threadIdx.x & 31;
    const int m16  = lane & 15;
    const int kk   = (lane >> 4) << 1;
    const int n0   = (blockIdx.x * 8 + wave) * 16;

    v8f zero = {0.f, 0.f, 0.f, 0.f, 0.f, 0.f, 0.f, 0.f};
    v8f acc[4];
#pragma unroll
    for (int r = 0; r < 4; ++r) acc[r] = zero;

#pragma unroll 4
    for (int k0 = 0; k0 < HID; k0 += 4) {
        v2f bf;
        bf.x = W2[(k0 + kk) * ODIM + n0 + m16];
        bf.y = W2[(k0 + kk + 1) * ODIM + n0 + m16];
#pragma unroll
        for (int r = 0; r < 4; ++r) {
            v2f a = *(const v2f*)(h + (r * 16 + m16) * HID + k0 + kk);
            acc[r] = wmma4_f32(a, bf, acc[r]);
        }
    }

    const float bv   = b2[n0 + m16];
    const int   row0 = (lane >> 4) * 8;
#pragma unroll
    for (int r = 0; r < 4; ++r) {
#pragma unroll
        for (int v = 0; v < 8; ++v) {
            y[(r * 16 + row0 + v) * ODIM + n0 + m16] = acc[r][v] + bv;
        }
    }
}

extern "C" void kernel_launch(void* const* d_in, const int* in_sizes, int n_in,
                              void* d_out, int out_size, void* d_ws, size_t ws_size,
                              hipStream_t stream) {
    const float* x   = (const float*)d_in[0];  // (512, 64, 256)
    const float* Wx  = (const float*)d_in[1];  // (256, 1024)
    const float* Wh  = (const float*)d_in[2];  // (1024, 1024)
    const float* b   = (const float*)d_in[3];  // (1024)
    const float* W2  = (const float*)d_in[4];  // (1024, 256)
    const float* b2  = (const float*)d_in[5];  // (256)
    float* yout      = (float*)d_out;          // (64, 256)

    // double-buffered hidden state in workspace
    float* h0 = (float*)d_ws;
    float* h1 = h0 + (size_t)BATCH * HID;

    // h(-1) = 0
    const int hn = BATCH * HID;
    rnn_zero_kernel<<<(hn + 255) / 256, 256, 0, stream>>>(h0, hn);

    // Steps t = 0 .. 510 (reference output uses H[-2] = h after step 510).
    // t even: read h0, write h1 ; t odd: read h1, write h0.
    for (int t = 0; t < T_STEPS - 1; ++t) {
        const float* hin = (t & 1) ? h1 : h0;
        float*      hout = (t & 1) ? h0 : h1;
        rnn_step_kernel<<<8, 256, 0, stream>>>(
            x + (size_t)t * BATCH * DIN, Wx, Wh, b, hin, hout);
    }
    // t = 510 is even -> final hidden state in h1
    rnn_out_kernel<<<2, 256, 0, stream>>>(h1, W2, b2, yout);
}